// ODEVAE_54692113547818
// MI455X (gfx1250) — compile-verified
//
#include <hip/hip_runtime.h>

typedef __attribute__((ext_vector_type(2))) float v2f;
typedef __attribute__((ext_vector_type(8))) float v8f;

#define DTC 0.1f
#define EPSF 1e-5f

// Output layout (floats): Xrec, qz0_m, qz0_v, zt, lhood, kl
#define XREC_OFF 0
#define QM_OFF   10035200
#define QV_OFF   10037248
#define ZT_OFF   10039296
#define LH_OFF   10141696
#define KL_OFF   10141697

__device__ __forceinline__ v8f wmma_f32_16x16x4(v2f a, v2f b, v8f c) {
  return __builtin_amdgcn_wmma_f32_16x16x4_f32(false, a, false, b, (short)0, c,
                                               false, false);
}

// ---------------- Kernel 1: encoder convs on frame 0, one block per sample ---
__global__ __launch_bounds__(256) void encoder_kernel(
    const float* __restrict__ X,
    const float* __restrict__ w1, const float* __restrict__ b1,
    const float* __restrict__ w2, const float* __restrict__ b2,
    const float* __restrict__ w3, const float* __restrict__ b3,
    float* __restrict__ h_out) {
  __shared__ float sx[784];
  __shared__ float sh1[8 * 196];
  __shared__ float sh2[16 * 49];
  __shared__ float sw1[200];
  __shared__ float sw2[3200];
  __shared__ float sw3[12800];
  __shared__ float sb1[8], sb2[16], sb3[32];
  int n = blockIdx.x, tid = threadIdx.x;
  const float* xp = X + (size_t)n * 50 * 784;  // frame t=0
  for (int i = tid; i < 784; i += 256) sx[i] = xp[i];
  for (int i = tid; i < 200; i += 256) sw1[i] = w1[i];
  for (int i = tid; i < 3200; i += 256) sw2[i] = w2[i];
  for (int i = tid; i < 12800; i += 256) sw3[i] = w3[i];
  if (tid < 8) sb1[tid] = b1[tid];
  if (tid < 16) sb2[tid] = b2[tid];
  if (tid < 32) sb3[tid] = b3[tid];
  __syncthreads();
  // conv1: 1->8, 28->14
  for (int i = tid; i < 8 * 196; i += 256) {
    int oc = i / 196, p = i % 196, oy = p / 14, ox = p % 14;
    float acc = sb1[oc];
    for (int kh = 0; kh < 5; ++kh) {
      int iy = oy * 2 + kh - 2;
      if (iy < 0 || iy > 27) continue;
      for (int kw = 0; kw < 5; ++kw) {
        int ix = ox * 2 + kw - 2;
        if (ix < 0 || ix > 27) continue;
        acc += sw1[oc * 25 + kh * 5 + kw] * sx[iy * 28 + ix];
      }
    }
    sh1[i] = fmaxf(acc, 0.f);
  }
  __syncthreads();
  // conv2: 8->16, 14->7
  for (int i = tid; i < 784; i += 256) {
    int oc = i / 49, p = i % 49, oy = p / 7, ox = p % 7;
    float acc = sb2[oc];
    for (int kh = 0; kh < 5; ++kh) {
      int iy = oy * 2 + kh - 2;
      if (iy < 0 || iy > 13) continue;
      for (int kw = 0; kw < 5; ++kw) {
        int ix = ox * 2 + kw - 2;
        if (ix < 0 || ix > 13) continue;
        int tap = kh * 5 + kw;
        const float* hp = &sh1[iy * 14 + ix];
        const float* wp = &sw2[(oc * 8) * 25 + tap];
        float s = 0.f;
        for (int ic = 0; ic < 8; ++ic) s += hp[ic * 196] * wp[ic * 25];
        acc += s;
      }
    }
    sh2[i] = fmaxf(acc, 0.f);
  }
  __syncthreads();
  // conv3: 16->32, 7->4, write relu'd h to workspace
  for (int i = tid; i < 512; i += 256) {
    int oc = i / 16, p = i % 16, oy = p / 4, ox = p % 4;
    float acc = sb3[oc];
    for (int kh = 0; kh < 5; ++kh) {
      int iy = oy * 2 + kh - 2;
      if (iy < 0 || iy > 6) continue;
      for (int kw = 0; kw < 5; ++kw) {
        int ix = ox * 2 + kw - 2;
        if (ix < 0 || ix > 6) continue;
        int tap = kh * 5 + kw;
        const float* hp = &sh2[iy * 7 + ix];
        const float* wp = &sw3[(oc * 16) * 25 + tap];
        float s = 0.f;
        for (int ic = 0; ic < 16; ++ic) s += hp[ic * 49] * wp[ic * 25];
        acc += s;
      }
    }
    h_out[(size_t)n * 512 + i] = fmaxf(acc, 0.f);
  }
}

// --------- Kernel 2: fc1|fc2 fused WMMA GEMM + reparameterize (1 wave/tile) --
__global__ __launch_bounds__(32) void fc12_kernel(
    const float* __restrict__ h,
    const float* __restrict__ fc1_w, const float* __restrict__ fc1_b,
    const float* __restrict__ fc2_w, const float* __restrict__ fc2_b,
    const float* __restrict__ eps_z, float* __restrict__ out) {
  __shared__ float st[256];
  int tile = blockIdx.x, lane = threadIdx.x;
  int m16 = lane & 15, hi = lane >> 4;
  // Hoist per-lane base pointers out of the k-loop (loop-invariant lane->matrix
  // mapping; avoids per-iteration EXEC-mask divergence).
  const float* ap = h + (size_t)(tile * 16 + m16) * 512 + hi * 2;
  const float* bp = (m16 < 8) ? (fc1_w + m16 * 512 + hi * 2)
                              : (fc2_w + (m16 - 8) * 512 + hi * 2);
  v8f acc = {};
  for (int ks = 0; ks < 128; ++ks) {
    int kk = ks * 4;
    v2f a, b;
    a.x = ap[kk];
    a.y = ap[kk + 1];
    b.x = bp[kk];
    b.y = bp[kk + 1];
    acc = wmma_f32_16x16x4(a, b, acc);
  }
  for (int r = 0; r < 8; ++r) st[(r + hi * 8) * 16 + m16] = acc[r];
  __syncthreads();
  for (int idx = lane; idx < 128; idx += 32) {
    int r = idx >> 3, c = idx & 7;
    int row = tile * 16 + r;
    float qm = st[r * 16 + c] + fc1_b[c];
    float lv = st[r * 16 + 8 + c] + fc2_b[c];
    float v = (lv > 20.f) ? lv : log1pf(expf(lv));  // softplus
    float z0 = qm + eps_z[row * 8 + c] * v;
    out[QM_OFF + row * 8 + c] = qm;
    out[QV_OFF + row * 8 + c] = v;
    out[ZT_OFF + (row * 50 + 0) * 8 + c] = z0;
  }
}

// ------------- Kernel 3: RK4 ODE, 16 samples/block, WMMA MLP, LDS weights ----
__global__ __launch_bounds__(128) void ode_kernel(
    const float* __restrict__ fw1, const float* __restrict__ fb1,
    const float* __restrict__ fw2, const float* __restrict__ fb2,
    const float* __restrict__ fw3, const float* __restrict__ fb3,
    float* __restrict__ out) {
  __shared__ float w1t[8 * 112];     // B[k][n] layer1, n padded to 112
  __shared__ float w2t[100 * 112];   // B[k][n] layer2
  __shared__ float w3t[100 * 16];    // B[k][j] layer3, j padded to 16
  __shared__ float b1s[112], b2s[112], b3s[16];
  __shared__ float zs[128], vts[128], kst[4][128];
  __shared__ float a1s[16 * 112], a2s[16 * 112];
  int tid = threadIdx.x, wave = tid >> 5, lane = tid & 31;
  int m16 = lane & 15, hi = lane >> 4;
  int n0 = blockIdx.x * 16;
  for (int i = tid; i < 8 * 112; i += 128) {
    int k = i / 112, n = i % 112;
    w1t[i] = (n < 100) ? fw1[n * 8 + k] : 0.f;
  }
  for (int i = tid; i < 100 * 112; i += 128) {
    int k = i / 112, n = i % 112;
    w2t[i] = (n < 100) ? fw2[n * 100 + k] : 0.f;
  }
  for (int i = tid; i < 100 * 16; i += 128) {
    int k = i / 16, n = i % 16;
    w3t[i] = (n < 8) ? fw3[n * 100 + k] : 0.f;
  }
  for (int i = tid; i < 112; i += 128) {
    b1s[i] = (i < 100) ? fb1[i] : 0.f;
    b2s[i] = (i < 100) ? fb2[i] : 0.f;
  }
  if (tid < 16) b3s[tid] = (tid < 8) ? fb3[tid] : 0.f;
  {
    int m = tid >> 3, j = tid & 7;
    zs[tid] = out[ZT_OFF + ((n0 + m) * 50) * 8 + j];
  }
  __syncthreads();
  const float coef[4] = {0.f, 0.5f, 0.5f, 1.f};
  for (int t = 1; t < 50; ++t) {
    for (int e = 0; e < 4; ++e) {
      {
        float v = zs[tid];
        if (e > 0) v += coef[e] * DTC * kst[e - 1][tid];
        vts[tid] = v;
      }
      __syncthreads();
      // layer1: [16,8]@[8,112]
      for (int nt = wave; nt < 7; nt += 4) {
        v8f acc = {};
        for (int ks = 0; ks < 2; ++ks) {
          int kk = ks * 4 + hi * 2;
          v2f a, b;
          a.x = vts[m16 * 8 + kk];
          a.y = vts[m16 * 8 + kk + 1];
          int col = nt * 16 + m16;
          b.x = w1t[kk * 112 + col];
          b.y = w1t[(kk + 1) * 112 + col];
          acc = wmma_f32_16x16x4(a, b, acc);
        }
        for (int r = 0; r < 8; ++r) {
          int mr = r + hi * 8, col = nt * 16 + m16;
          a1s[mr * 112 + col] = fmaxf(acc[r] + b1s[col], 0.f);
        }
      }
      __syncthreads();
      // layer2: [16,100]@[100,112]
      for (int nt = wave; nt < 7; nt += 4) {
        const float* arow = &a1s[m16 * 112 + hi * 2];
        const float* bcol = &w2t[hi * 2 * 112 + nt * 16 + m16];
        v8f acc = {};
        for (int ks = 0; ks < 25; ++ks) {
          v2f a, b;
          a.x = arow[ks * 4];
          a.y = arow[ks * 4 + 1];
          b.x = bcol[ks * 4 * 112];
          b.y = bcol[ks * 4 * 112 + 112];
          acc = wmma_f32_16x16x4(a, b, acc);
        }
        for (int r = 0; r < 8; ++r) {
          int mr = r + hi * 8, col = nt * 16 + m16;
          a2s[mr * 112 + col] = fmaxf(acc[r] + b2s[col], 0.f);
        }
      }
      __syncthreads();
      // layer3: [16,100]@[100,16] (cols 8..15 are zero pad), wave 0 only
      if (wave == 0) {
        const float* arow = &a2s[m16 * 112 + hi * 2];
        const float* bcol = &w3t[hi * 2 * 16 + m16];
        v8f acc = {};
        for (int ks = 0; ks < 25; ++ks) {
          v2f a, b;
          a.x = arow[ks * 4];
          a.y = arow[ks * 4 + 1];
          b.x = bcol[ks * 4 * 16];
          b.y = bcol[ks * 4 * 16 + 16];
          acc = wmma_f32_16x16x4(a, b, acc);
        }
        if (m16 < 8) {
          for (int r = 0; r < 8; ++r) {
            int mr = r + hi * 8;
            kst[e][mr * 8 + m16] = acc[r] + b3s[m16];
          }
        }
      }
      __syncthreads();
    }
    {
      int m = tid >> 3, j = tid & 7;
      float zn = zs[tid] + (DTC / 6.f) * (kst[0][tid] + 2.f * kst[1][tid] +
                                          2.f * kst[2][tid] + kst[3][tid]);
      zs[tid] = zn;
      out[ZT_OFF + ((n0 + m) * 50 + t) * 8 + j] = zn;
    }
    __syncthreads();
  }
}

// ---- Kernel 4: decoder, 16 images/block: fc3+deconv1 WMMA, deconv2/3 VALU ---
__global__ __launch_bounds__(128) void decoder_kernel(
    const float* __restrict__ X,
    const float* __restrict__ fc3_w, const float* __restrict__ fc3_b,
    const float* __restrict__ w1, const float* __restrict__ db1,
    const float* __restrict__ w2, const float* __restrict__ db2,
    const float* __restrict__ w3, const float* __restrict__ db3,
    float* __restrict__ out, float* __restrict__ partials) {
  __shared__ float sw1t[25 * 512];  // [tap][ic(32)][oc(16)]  B for deconv1
  __shared__ float sw2[3200];       // dec_w2 raw [8][16][5][5]
  __shared__ float sw3[200];        // dec_w3 raw [1][8][5][5]
  __shared__ float sfc3b[512];
  __shared__ float sdb1[16], sdb2[8], sdb3[1];
  __shared__ float szt[128];
  __shared__ float sgy[8192];       // fc3 output g [16][512]; later y2 chunk [4][8][196]
  __shared__ float sy1[16 * 784];   // y1 [img][oc(16)][49]
  __shared__ float sred[128];
  int tid = threadIdx.x, wave = tid >> 5, lane = tid & 31;
  int m16 = lane & 15, hi = lane >> 4;
  int i0 = blockIdx.x * 16;  // first image index (n*50+t flat)
  // Prefetch the X tile consumed by the likelihood phase (overlaps with WMMA).
  for (int i = tid * 32; i < 16 * 784; i += 128 * 32)
    __builtin_prefetch(&X[(size_t)i0 * 784 + i], 0, 1);
  for (int i = tid; i < 12800; i += 128) {
    int tap = i >> 9, r = i & 511, ic = r >> 4, oc = r & 15;
    sw1t[i] = w1[(oc * 32 + ic) * 25 + tap];
  }
  for (int i = tid; i < 3200; i += 128) sw2[i] = w2[i];
  for (int i = tid; i < 200; i += 128) sw3[i] = w3[i];
  for (int i = tid; i < 512; i += 128) sfc3b[i] = fc3_b[i];
  if (tid < 16) sdb1[tid] = db1[tid];
  if (tid < 8) sdb2[tid] = db2[tid];
  if (tid == 0) sdb3[0] = db3[0];
  szt[tid] = out[ZT_OFF + i0 * 8 + tid];  // zt flat == per-image 8 floats
  __syncthreads();
  // fc3: g[16][512] = zt[16][8] @ fc3_w.T, WMMA (K=8 -> 2 k-steps)
  for (int nt = wave; nt < 32; nt += 4) {
    const float* bp = fc3_w + (nt * 16 + m16) * 8 + hi * 2;
    v8f acc = {};
    for (int ks = 0; ks < 2; ++ks) {
      int kk = ks * 4;
      v2f a, b;
      a.x = szt[m16 * 8 + kk + hi * 2];
      a.y = szt[m16 * 8 + kk + hi * 2 + 1];
      b.x = bp[kk];
      b.y = bp[kk + 1];
      acc = wmma_f32_16x16x4(a, b, acc);
    }
    for (int r = 0; r < 8; ++r) {
      int mr = r + hi * 8, col = nt * 16 + m16;
      sgy[mr * 512 + col] = acc[r] + sfc3b[col];
    }
  }
  __syncthreads();
  // deconv1 implicit-GEMM WMMA: 32ic x 4x4 -> 16oc x 7x7, lhs_dilation 2, pad 2
  for (int pos = wave; pos < 49; pos += 4) {
    int oy = pos / 7, ox = pos % 7;
    v8f acc = {};
    for (int kh = 0; kh < 5; ++kh) {
      int jy = oy + kh - 2;
      if (jy < 0 || jy > 6 || (jy & 1)) continue;
      int iy = jy >> 1;
      for (int kw = 0; kw < 5; ++kw) {
        int jx = ox + kw - 2;
        if (jx < 0 || jx > 6 || (jx & 1)) continue;
        int ix = jx >> 1;
        int tap = kh * 5 + kw;
        int sp = iy * 4 + ix;
        const float* ap = &sgy[m16 * 512 + hi * 2 * 16 + sp];
        const float* bp = &sw1t[tap * 512 + hi * 2 * 16 + m16];
        for (int ks = 0; ks < 8; ++ks) {  // K = 32 input channels
          v2f a, b;
          a.x = ap[ks * 64];
          a.y = ap[ks * 64 + 16];
          b.x = bp[ks * 64];
          b.y = bp[ks * 64 + 16];
          acc = wmma_f32_16x16x4(a, b, acc);
        }
      }
    }
    for (int r = 0; r < 8; ++r) {
      int mr = r + hi * 8;
      sy1[mr * 784 + m16 * 49 + pos] = fmaxf(acc[r] + sdb1[m16], 0.f);
    }
  }
  __syncthreads();
  float lh = 0.f;
  for (int c = 0; c < 4; ++c) {  // 4-image chunks; y2 reuses fc3's buffer
    // deconv2: 16ic x 7x7 -> 8oc x 14x14 (out_pad 1)
    for (int idx = tid; idx < 6272; idx += 128) {
      int il = idx / 1568, r = idx % 1568, oc = r / 196, p = r % 196;
      int oy = p / 14, ox = p % 14;
      int img = c * 4 + il;
      float acc = sdb2[oc];
      for (int kh = 0; kh < 5; ++kh) {
        int jy = oy + kh - 2;
        if (jy < 0 || jy > 12 || (jy & 1)) continue;
        int iy = jy >> 1;
        for (int kw = 0; kw < 5; ++kw) {
          int jx = ox + kw - 2;
          if (jx < 0 || jx > 12 || (jx & 1)) continue;
          int ix = jx >> 1;
          int tap = kh * 5 + kw;
          const float* y1p = &sy1[img * 784 + iy * 7 + ix];
          const float* wp = &sw2[(oc * 16) * 25 + tap];
          float s = 0.f;
          for (int ic = 0; ic < 16; ++ic) s += y1p[ic * 49] * wp[ic * 25];
          acc += s;
        }
      }
      sgy[idx] = fmaxf(acc, 0.f);
    }
    __syncthreads();
    // deconv3 + sigmoid + write Xrec + Bernoulli log-likelihood
    for (int idx = tid; idx < 3136; idx += 128) {
      int il = idx / 784, p = idx % 784;
      int oy = p / 28, ox = p % 28;
      int img = i0 + c * 4 + il;
      float acc = sdb3[0];
      for (int kh = 0; kh < 5; ++kh) {
        int jy = oy + kh - 2;
        if (jy < 0 || jy > 26 || (jy & 1)) continue;
        int iy = jy >> 1;
        for (int kw = 0; kw < 5; ++kw) {
          int jx = ox + kw - 2;
          if (jx < 0 || jx > 26 || (jx & 1)) continue;
          int ix = jx >> 1;
          int tap = kh * 5 + kw;
          const float* y2p = &sgy[il * 1568 + iy * 14 + ix];
          const float* wp = &sw3[tap];
          float s = 0.f;
          for (int ic = 0; ic < 8; ++ic) s += y2p[ic * 196] * wp[ic * 25];
          acc += s;
        }
      }
      float xr = 1.f / (1.f + expf(-acc));
      out[XREC_OFF + (size_t)img * 784 + p] = xr;
      float xv = X[(size_t)img * 784 + p];
      lh += logf(EPSF + xr) * xv + logf(EPSF + 1.f - xr) * (1.f - xv);
    }
    __syncthreads();
  }
  sred[tid] = lh;
  __syncthreads();
  for (int s = 64; s > 0; s >>= 1) {
    if (tid < s) sred[tid] += sred[tid + s];
    __syncthreads();
  }
  if (tid == 0) partials[blockIdx.x] = sred[0];
}

// ---------------- Kernel 5: KL scalar ---------------------------------------
__global__ __launch_bounds__(256) void kl_kernel(float* __restrict__ out) {
  __shared__ float sred[256];
  int tid = threadIdx.x;
  float s = 0.f;
  for (int i = tid; i < 2048; i += 256) {
    float m = out[QM_OFF + i], v = out[QV_OFF + i];
    s += -logf(v) + 0.5f * (v * v + m * m) - 0.5f;
  }
  sred[tid] = s;
  __syncthreads();
  for (int st = 128; st > 0; st >>= 1) {
    if (tid < st) sred[tid] += sred[tid + st];
    __syncthreads();
  }
  if (tid == 0) out[KL_OFF] = sred[0] / 2048.f;
}

// ---------------- Kernel 6: lhood scalar ------------------------------------
__global__ __launch_bounds__(256) void lh_kernel(const float* __restrict__ partials,
                                                 float* __restrict__ out) {
  __shared__ float sred[256];
  int tid = threadIdx.x;
  float s = 0.f;
  for (int i = tid; i < 800; i += 256) s += partials[i];
  sred[tid] = s;
  __syncthreads();
  for (int st = 128; st > 0; st >>= 1) {
    if (tid < st) sred[tid] += sred[tid + st];
    __syncthreads();
  }
  if (tid == 0) out[LH_OFF] = sred[0] / 256.f;
}

extern "C" void kernel_launch(void* const* d_in, const int* in_sizes, int n_in,
                              void* d_out, int out_size, void* d_ws, size_t ws_size,
                              hipStream_t stream) {
  const float* X      = (const float*)d_in[0];
  const float* eps_z  = (const float*)d_in[1];
  const float* enc_w1 = (const float*)d_in[2];
  const float* enc_b1 = (const float*)d_in[3];
  const float* enc_w2 = (const float*)d_in[4];
  const float* enc_b2 = (const float*)d_in[5];
  const float* enc_w3 = (const float*)d_in[6];
  const float* enc_b3 = (const float*)d_in[7];
  const float* fc1_w  = (const float*)d_in[8];
  const float* fc1_b  = (const float*)d_in[9];
  const float* fc2_w  = (const float*)d_in[10];
  const float* fc2_b  = (const float*)d_in[11];
  const float* f_w1   = (const float*)d_in[12];
  const float* f_b1   = (const float*)d_in[13];
  const float* f_w2   = (const float*)d_in[14];
  const float* f_b2   = (const float*)d_in[15];
  const float* f_w3   = (const float*)d_in[16];
  const float* f_b3   = (const float*)d_in[17];
  const float* fc3_w  = (const float*)d_in[18];
  const float* fc3_b  = (const float*)d_in[19];
  const float* dec_w1 = (const float*)d_in[20];
  const float* dec_b1 = (const float*)d_in[21];
  const float* dec_w2 = (const float*)d_in[22];
  const float* dec_b2 = (const float*)d_in[23];
  const float* dec_w3 = (const float*)d_in[24];
  const float* dec_b3 = (const float*)d_in[25];
  (void)in_sizes; (void)n_in; (void)out_size; (void)ws_size;

  float* out = (float*)d_out;
  float* wsF = (float*)d_ws;
  float* h        = wsF;            // 256*512 floats
  float* partials = wsF + 131072;   // 800 floats

  encoder_kernel<<<256, 256, 0, stream>>>(X, enc_w1, enc_b1, enc_w2, enc_b2,
                                          enc_w3, enc_b3, h);
  fc12_kernel<<<16, 32, 0, stream>>>(h, fc1_w, fc1_b, fc2_w, fc2_b, eps_z, out);
  ode_kernel<<<16, 128, 0, stream>>>(f_w1, f_b1, f_w2, f_b2, f_w3, f_b3, out);
  decoder_kernel<<<800, 128, 0, stream>>>(X, fc3_w, fc3_b, dec_w1, dec_b1,
                                          dec_w2, dec_b2, dec_w3, dec_b3, out,
                                          partials);
  kl_kernel<<<1, 256, 0, stream>>>(out);
  lh_kernel<<<1, 256, 0, stream>>>(partials, out);
}